// SsimLoss_5196910428170
// MI455X (gfx1250) — compile-verified
//
#include <hip/hip_runtime.h>
#include <math.h>

typedef __attribute__((ext_vector_type(2))) float v2f;
typedef __attribute__((ext_vector_type(8))) float v8f;

#define WS        11
#define HALF_WS   5
#define TWO_SIG2  4.5f           // 2 * 1.5^2
#define TILE      16
#define PATCH     28             // 16 + (11-1) = 26, padded to 28 for K-chunks
#define NQ        7              // ceil(28/4) K-chunks of the f32 WMMA (K=4)
#define BATCH     32
#define H_IN      480
#define W_IN      640
#define H_OUT     (H_IN - WS + 1)           // 470
#define W_OUT     (W_IN - WS + 1)           // 630
#define TILES_Y   ((H_OUT + TILE - 1) / TILE)   // 30
#define TILES_X   ((W_OUT + TILE - 1) / TILE)   // 40
#define TILES_IMG (TILES_Y * TILES_X)           // 1200
#define NTILES    (TILES_IMG * BATCH)           // 38400
#define WPB       8                              // waves per block (256 threads)
#define NBLOCKS   (NTILES / WPB)                 // 4800 exactly
#define SSIM_C1   1.0f           // (0.01*100)^2
#define SSIM_C2   9.0f           // (0.03*100)^2

__device__ __forceinline__ float fieldval(int f, float a, float b) {
    switch (f) {
    case 0:  return a;
    case 1:  return b;
    case 2:  return a * a;
    case 3:  return b * b;
    default: return a * b;
    }
}

__global__ __launch_bounds__(256) void ssim_tiles(const float* __restrict__ img1,
                                                  const float* __restrict__ img2,
                                                  float* __restrict__ partial)
{
    __shared__ float gw[16];
    __shared__ float pA[WPB][PATCH][PATCH];
    __shared__ float pB[WPB][PATCH][PATCH];
    __shared__ float hb[WPB][PATCH][TILE];
    __shared__ float wsum[WPB];

    const int tid  = threadIdx.x;
    const int wave = tid >> 5;
    const int lane = tid & 31;
    const int half = lane >> 4;      // which 16-lane half (selects K pair / M+8)
    const int ln   = lane & 15;      // N (or M) index within the tile

    // --- normalized Gaussian window, computed once per block ---
    if (tid < 16) {
        float s = 0.f;
        #pragma unroll
        for (int k = 0; k < WS; ++k) {
            float d = (float)(k - HALF_WS);
            s += expf(-d * d / TWO_SIG2);
        }
        float v = 0.f;
        if (tid < WS) {
            float d = (float)(tid - HALF_WS);
            v = expf(-d * d / TWO_SIG2) / s;
        }
        gw[tid] = v;
    }

    // --- tile coordinates (one 16x16 output tile per wave) ---
    const int tileIdx = blockIdx.x * WPB + wave;
    const int imgIdx  = tileIdx / TILES_IMG;
    const int rem     = tileIdx % TILES_IMG;
    const int ty      = rem / TILES_X;
    const int tx      = rem % TILES_X;
    const int r0      = ty * TILE;
    const int c0      = tx * TILE;
    const long long base = (long long)imgIdx * (H_IN * W_IN);

    // --- load 28x28 patches (border-clamped; clamped values only ever meet
    //     zero band coefficients or get masked in the reduction) ---
    for (int idx = lane; idx < PATCH * PATCH; idx += 32) {
        int pr = idx / PATCH, pc = idx % PATCH;
        int gr = min(r0 + pr, H_IN - 1);
        int gc = min(c0 + pc, W_IN - 1);
        long long off = base + (long long)gr * W_IN + gc;
        pA[wave][pr][pc] = img1[off];
        pB[wave][pr][pc] = img2[off];
    }
    __syncthreads();   // gw + patches visible

    // --- banded Gaussian fragments: value at (K=m, other=ln) is g[m-ln].
    //     Same lane layout for A(16x4) and B(4x16): K = 2*half + v, other = ln,
    //     so one fragment serves as B (horizontal) and A (vertical). ---
    v2f bnd[NQ];
    #pragma unroll
    for (int q = 0; q < NQ; ++q) {
        #pragma unroll
        for (int v = 0; v < 2; ++v) {
            int m  = 4 * q + 2 * half + v;
            int u  = m - ln;
            int uc = (u < 0) ? 0 : ((u >= WS) ? (WS - 1) : u);
            bnd[q][v] = (u >= 0 && u < WS) ? gw[uc] : 0.f;
        }
    }

    const int row0 = ln;                       // rows 0..15
    const int row1 = min(16 + ln, PATCH - 1);  // rows 16..27 (28..31 are dead)

    v8f acc[5];
    #pragma unroll
    for (int f = 0; f < 5; ++f) {
        // ---- horizontal pass: H[r][j] = sum_k g[k] * field[r][j+k] ----
        v8f h0 = {0.f, 0.f, 0.f, 0.f, 0.f, 0.f, 0.f, 0.f};
        v8f h1 = {0.f, 0.f, 0.f, 0.f, 0.f, 0.f, 0.f, 0.f};
        #pragma unroll
        for (int q = 0; q < NQ; ++q) {
            v2f a0, a1;
            #pragma unroll
            for (int v = 0; v < 2; ++v) {
                int m = 4 * q + 2 * half + v;          // patch column
                float x0 = pA[wave][row0][m], y0 = pB[wave][row0][m];
                float x1 = pA[wave][row1][m], y1 = pB[wave][row1][m];
                a0[v] = fieldval(f, x0, y0);
                a1[v] = fieldval(f, x1, y1);
            }
            h0 = __builtin_amdgcn_wmma_f32_16x16x4_f32(false, a0, false, bnd[q],
                                                       (short)0, h0, false, false);
            h1 = __builtin_amdgcn_wmma_f32_16x16x4_f32(false, a1, false, bnd[q],
                                                       (short)0, h1, false, false);
        }
        // Stash intermediate. Rows >= 26 only ever meet zero band coefficients,
        // so their values are dead as long as they are FINITE (0*Inf -> NaN).
        // Clamp the destination row instead of branching: rows 28..31 collapse
        // onto row 27 (finite garbage, coefficient 0) -> no exec-mask dance.
        #pragma unroll
        for (int v = 0; v < 8; ++v) {
            int r   = v + 8 * half;                    // 0..15
            int r2c = min(16 + r, PATCH - 1);          // 16..27 (clamped)
            hb[wave][r][ln]   = h0[v];
            hb[wave][r2c][ln] = h1[v];
        }
        // ---- vertical pass: V[i][j] = sum_k g[k] * H[i+k][j] ----
        v8f vac = {0.f, 0.f, 0.f, 0.f, 0.f, 0.f, 0.f, 0.f};
        #pragma unroll
        for (int q = 0; q < NQ; ++q) {
            v2f bfr;
            #pragma unroll
            for (int v = 0; v < 2; ++v) {
                int m = 4 * q + 2 * half + v;          // intermediate row
                bfr[v] = hb[wave][m][ln];
            }
            vac = __builtin_amdgcn_wmma_f32_16x16x4_f32(false, bnd[q], false, bfr,
                                                        (short)0, vac, false, false);
        }
        acc[f] = vac;
    }

    // --- pointwise SSIM + masked partial sum ---
    float lsum = 0.f;
    #pragma unroll
    for (int v = 0; v < 8; ++v) {
        int orow = r0 + v + 8 * half;   // C/D layout: M = v + 8*(lane>=16)
        int ocol = c0 + ln;
        if (orow < H_OUT && ocol < W_OUT) {
            float mu1 = acc[0][v], mu2 = acc[1][v];
            float mu1s = mu1 * mu1, mu2s = mu2 * mu2, mu12 = mu1 * mu2;
            float s1  = acc[2][v] - mu1s;
            float s2  = acc[3][v] - mu2s;
            float s12 = acc[4][v] - mu12;
            float num = (2.f * mu12 + SSIM_C1) * (2.f * s12 + SSIM_C2);
            float den = (mu1s + mu2s + SSIM_C1) * (s1 + s2 + SSIM_C2);
            lsum += num / den;
        }
    }
    #pragma unroll
    for (int off = 16; off > 0; off >>= 1)
        lsum += __shfl_down(lsum, off, 32);
    if (lane == 0) wsum[wave] = lsum;
    __syncthreads();
    if (tid == 0) {
        float s = 0.f;
        #pragma unroll
        for (int w = 0; w < WPB; ++w) s += wsum[w];
        partial[blockIdx.x] = s;
    }
}

__global__ __launch_bounds__(256) void ssim_reduce(const float* __restrict__ partial,
                                                   float* __restrict__ out)
{
    __shared__ float sm[256];
    const int tid = threadIdx.x;
    float a = 0.f;
    for (int i = tid; i < NBLOCKS; i += 256) a += partial[i];   // fixed order
    sm[tid] = a;
    __syncthreads();
    for (int s = 128; s > 0; s >>= 1) {
        if (tid < s) sm[tid] += sm[tid + s];
        __syncthreads();
    }
    if (tid == 0) {
        float mean = sm[0] / (float)((long long)BATCH * H_OUT * W_OUT);
        out[0] = (1.f - mean) * 0.5f;
    }
}

extern "C" void kernel_launch(void* const* d_in, const int* in_sizes, int n_in,
                              void* d_out, int out_size, void* d_ws, size_t ws_size,
                              hipStream_t stream)
{
    (void)in_sizes; (void)n_in; (void)out_size; (void)ws_size;
    const float* img1 = (const float*)d_in[0];
    const float* img2 = (const float*)d_in[1];
    float* partial = (float*)d_ws;             // NBLOCKS floats (19.2 KB)
    float* out = (float*)d_out;

    ssim_tiles<<<NBLOCKS, 256, 0, stream>>>(img1, img2, partial);
    ssim_reduce<<<1, 256, 0, stream>>>(partial, out);
}